// SingleNodeAttentionLayer_54434415509715
// MI455X (gfx1250) — compile-verified
//
#include <hip/hip_runtime.h>
#include <math.h>

typedef __attribute__((ext_vector_type(2))) float v2f;
typedef __attribute__((ext_vector_type(8))) float v8f;

#define IN_F   128
#define TWO_F  128          // 2*OUT_F
#define OUT_F  64
#define NHEADS 4
#define NODES  2048
#define NGROUPS 64          // 2 * 32
#define BTHREADS 1024       // 32 waves per group-block
#define NSLICES  (BTHREADS / IN_F)   // 8 node-range slices for stage 3

// ---------------------------------------------------------------------------
// Kernel A: vcat[h][i] = sum_j W_h[i][j] * a_h[j]   (since (xW)a == x(Wa))
// 1 block x 512 threads: thread t -> head t>>7, row t&127
// ---------------------------------------------------------------------------
__global__ void __launch_bounds__(512) gat_prep(
    const float* __restrict__ W1, const float* __restrict__ a1,
    const float* __restrict__ W2, const float* __restrict__ a2,
    const float* __restrict__ W3, const float* __restrict__ a3,
    const float* __restrict__ W4, const float* __restrict__ a4,
    float* __restrict__ vcat)
{
  const int t = threadIdx.x;
  const int h = t >> 7;
  const int i = t & 127;
  const float* W = (h == 0) ? W1 : (h == 1) ? W2 : (h == 2) ? W3 : W4;
  const float* a = (h == 0) ? a1 : (h == 1) ? a2 : (h == 2) ? a3 : a4;
  float s = 0.f;
#pragma unroll 4
  for (int j = 0; j < TWO_F; ++j) s += W[i * TWO_F + j] * a[j];
  vcat[h * IN_F + i] = s;
}

// ---------------------------------------------------------------------------
// Kernel B: one 1024-thread block per (b,g) group of 2048 nodes.
//   Stage 1: scores s[n][h] = x_n . v_h  via V_WMMA_F32_16X16X4_F32
//            (B matrix = vcat zero-padded to 16 columns)
//   Stage 2: softmax over n (per head); wave shfl-reduce + small LDS tree
//   Stage 3: y_h = (sum_n e[n][h] * x_n) / sum_n e[n][h]
//   Stage 4: pooled_h = y_h @ W_h  -> multi[g][h*128 + j]
// ---------------------------------------------------------------------------
__global__ void __launch_bounds__(BTHREADS) gat_group(
    const float* __restrict__ X,      // [64,2048,128]
    const float* __restrict__ vcat,   // [4,128]
    const float* __restrict__ W1, const float* __restrict__ W2,
    const float* __restrict__ W3, const float* __restrict__ W4,
    float* __restrict__ multi)        // [64, 512]
{
  __shared__ float vc[NHEADS * IN_F];               // 2 KB
  __shared__ float s_lds[NODES * NHEADS];           // 32 KB: scores -> exp weights
  __shared__ float red[32 * NHEADS];                // 512 B: cross-wave reduce
  __shared__ float yred[NSLICES * NHEADS * IN_F];   // 16 KB: partial y
  __shared__ float y_lds[NHEADS * IN_F];            // 2 KB: pooled features

  const int t    = threadIdx.x;
  const int g    = blockIdx.x;
  const int wave = t >> 5;
  const int lane = t & 31;
  const float* __restrict__ Xg = X + (size_t)g * NODES * IN_F;

  if (t < NHEADS * IN_F) vc[t] = vcat[t];
  __syncthreads();

  // ---- Stage 1: WMMA scores --------------------------------------------
  {
    const int mrow = lane & 15;            // M row within tile
    const int koff = (lane >> 4) * 2;      // lanes 16-31 hold K+2,K+3
    const int head = lane & 15;            // N column == head index
    const int hidx = (head < NHEADS) ? head : 0;
    const float bmask = (head < NHEADS) ? 1.f : 0.f;

    for (int tile = wave; tile < NODES / 16; tile += 32) {   // 4 tiles / wave
      const float* __restrict__ rowp = Xg + (size_t)(tile * 16 + mrow) * IN_F + koff;
      v8f c = {};
#pragma unroll 8
      for (int k0 = 0; k0 < IN_F; k0 += 4) {
        float2 av = *(const float2*)(rowp + k0);
        v2f a; a.x = av.x; a.y = av.y;
        const int kb = k0 + koff;
        v2f b; b.x = vc[hidx * IN_F + kb] * bmask;
               b.y = vc[hidx * IN_F + kb + 1] * bmask;
        c = __builtin_amdgcn_wmma_f32_16x16x4_f32(
                false, a, false, b, (short)0, c, false, false);
      }
      const int rbase = (lane >> 4) * 8;   // D: VGPR r -> M=r (lo) / M=r+8 (hi)
      if (head < NHEADS) {
#pragma unroll
        for (int r = 0; r < 8; ++r)
          s_lds[(tile * 16 + rbase + r) * NHEADS + head] = c[r];
      }
    }
  }
  __syncthreads();

  // ---- Stage 2: softmax over 2048 nodes, per head ----------------------
  float lmax[NHEADS];
#pragma unroll
  for (int h = 0; h < NHEADS; ++h) lmax[h] = -3.4e38f;
  for (int n = t; n < NODES; n += BTHREADS) {                // 2 iterations
#pragma unroll
    for (int h = 0; h < NHEADS; ++h)
      lmax[h] = fmaxf(lmax[h], s_lds[n * NHEADS + h]);
  }
#pragma unroll
  for (int h = 0; h < NHEADS; ++h) {
#pragma unroll
    for (int off = 16; off > 0; off >>= 1)
      lmax[h] = fmaxf(lmax[h], __shfl_xor(lmax[h], off, 32));
  }
  if (lane == 0) {
#pragma unroll
    for (int h = 0; h < NHEADS; ++h) red[wave * NHEADS + h] = lmax[h];
  }
  __syncthreads();
  for (int stride = 16; stride > 0; stride >>= 1) {          // 32 waves -> 1
    if (t < stride * NHEADS) {
      const int w = t >> 2, h = t & 3;
      red[w * NHEADS + h] = fmaxf(red[w * NHEADS + h],
                                  red[(w + stride) * NHEADS + h]);
    }
    __syncthreads();
  }
  float gmax[NHEADS];
#pragma unroll
  for (int h = 0; h < NHEADS; ++h) gmax[h] = red[h];
  __syncthreads();

  float lsum[NHEADS] = {0.f, 0.f, 0.f, 0.f};
  for (int n = t; n < NODES; n += BTHREADS) {
#pragma unroll
    for (int h = 0; h < NHEADS; ++h) {
      float e = __expf(s_lds[n * NHEADS + h] - gmax[h]);
      s_lds[n * NHEADS + h] = e;
      lsum[h] += e;
    }
  }
#pragma unroll
  for (int h = 0; h < NHEADS; ++h) {
#pragma unroll
    for (int off = 16; off > 0; off >>= 1)
      lsum[h] += __shfl_xor(lsum[h], off, 32);
  }
  if (lane == 0) {
#pragma unroll
    for (int h = 0; h < NHEADS; ++h) red[wave * NHEADS + h] = lsum[h];
  }
  __syncthreads();
  for (int stride = 16; stride > 0; stride >>= 1) {
    if (t < stride * NHEADS) {
      const int w = t >> 2, h = t & 3;
      red[w * NHEADS + h] += red[(w + stride) * NHEADS + h];
    }
    __syncthreads();
  }
  // red[h] now holds sum_n exp(...) for head h; left untouched below.

  // ---- Stage 3: y_h = (e^T X) / sum ------------------------------------
  {
    const int col   = t & 127;
    const int slice = t >> 7;                   // 8 slices x 256 nodes
    const int n0 = slice * (NODES / NSLICES);
    const int n1 = n0 + (NODES / NSLICES);
    float acc[NHEADS] = {0.f, 0.f, 0.f, 0.f};
#pragma unroll 4
    for (int n = n0; n < n1; ++n) {
      float xv = Xg[(size_t)n * IN_F + col];    // coalesced across lanes
#pragma unroll
      for (int h = 0; h < NHEADS; ++h)
        acc[h] += s_lds[n * NHEADS + h] * xv;
    }
#pragma unroll
    for (int h = 0; h < NHEADS; ++h)
      yred[(slice * NHEADS + h) * IN_F + col] = acc[h];
  }
  __syncthreads();
  if (t < IN_F) {
#pragma unroll
    for (int h = 0; h < NHEADS; ++h) {
      float s = 0.f;
#pragma unroll
      for (int sl = 0; sl < NSLICES; ++sl)
        s += yred[(sl * NHEADS + h) * IN_F + t];
      y_lds[h * IN_F + t] = s * (1.f / red[h]);
    }
  }
  __syncthreads();

  // ---- Stage 4: pooled_h = y_h @ W_h -----------------------------------
  if (t < NHEADS * TWO_F) {                     // 512 outputs
    const float* Ws[NHEADS] = {W1, W2, W3, W4};
    const int h = t >> 7;
    const int j = t & 127;
    const float* __restrict__ W = Ws[h];
    float acc = 0.f;
#pragma unroll 4
    for (int i = 0; i < IN_F; ++i)
      acc += y_lds[h * IN_F + i] * W[i * TWO_F + j];
    multi[(size_t)g * (NHEADS * TWO_F) + h * TWO_F + j] = acc;
  }
}

// ---------------------------------------------------------------------------
// Kernel C: out = elu(multi[64,512] @ Wo[512,64]) via WMMA f32 16x16x4.
// One block, 16 waves; wave w -> output tile (w>>2, w&3); 128 k-steps.
// ---------------------------------------------------------------------------
__global__ void __launch_bounds__(512) gat_out(
    const float* __restrict__ multi,   // [64,512]
    const float* __restrict__ Wo,      // [512,64]
    float* __restrict__ out)           // [64,64]
{
  const int t = threadIdx.x;
  const int wave = t >> 5;
  const int lane = t & 31;
  const int tileM = wave >> 2;
  const int tileN = wave & 3;
  const int mrow = lane & 15;
  const int koff = (lane >> 4) * 2;
  const int ncol = tileN * 16 + (lane & 15);
  const float* __restrict__ arow =
      multi + (size_t)(tileM * 16 + mrow) * (8 * OUT_F) + koff;

  v8f c = {};
#pragma unroll 8
  for (int k0 = 0; k0 < 8 * OUT_F; k0 += 4) {
    float2 av = *(const float2*)(arow + k0);
    v2f a; a.x = av.x; a.y = av.y;
    v2f b; b.x = Wo[(size_t)(k0 + koff) * OUT_F + ncol];
           b.y = Wo[(size_t)(k0 + koff + 1) * OUT_F + ncol];
    c = __builtin_amdgcn_wmma_f32_16x16x4_f32(
            false, a, false, b, (short)0, c, false, false);
  }
  const int rbase = (lane >> 4) * 8;
#pragma unroll
  for (int r = 0; r < 8; ++r) {
    float v = c[r];
    float e = (v > 0.f) ? v : (__expf(v) - 1.f);   // ELU, alpha=1
    out[(size_t)(tileM * 16 + rbase + r) * OUT_F + ncol] = e;
  }
}

// ---------------------------------------------------------------------------
extern "C" void kernel_launch(void* const* d_in, const int* in_sizes, int n_in,
                              void* d_out, int out_size, void* d_ws, size_t ws_size,
                              hipStream_t stream) {
  const float* features = (const float*)d_in[0];
  const float* W1 = (const float*)d_in[1]; const float* a1 = (const float*)d_in[2];
  const float* W2 = (const float*)d_in[3]; const float* a2 = (const float*)d_in[4];
  const float* W3 = (const float*)d_in[5]; const float* a3 = (const float*)d_in[6];
  const float* W4 = (const float*)d_in[7]; const float* a4 = (const float*)d_in[8];
  const float* Wo = (const float*)d_in[9];

  float* ws    = (float*)d_ws;
  float* vcat  = ws;            // 512 floats
  float* multi = ws + 512;      // 64*512 floats (128 KB)

  gat_prep<<<1, 512, 0, stream>>>(W1, a1, W2, a2, W3, a3, W4, a4, vcat);
  gat_group<<<NGROUPS, BTHREADS, 0, stream>>>(features, vcat, W1, W2, W3, W4, multi);
  gat_out<<<1, 512, 0, stream>>>(multi, Wo, (float*)d_out);
}